// GCN_35588099015570
// MI455X (gfx1250) — compile-verified
//
#include <hip/hip_runtime.h>

typedef __attribute__((ext_vector_type(2))) float v2f;
typedef __attribute__((ext_vector_type(8))) float v8f;

#define HID 256
#define BN_EPS 1e-5f

// ---------------- degree / norm ----------------
__global__ void k_fill1(float* p, int n) {
    int i = blockIdx.x * blockDim.x + threadIdx.x;
    if (i < n) p[i] = 1.0f;            // self-loop contributes 1 to degree
}

__global__ void k_deg_edges(const long long* __restrict__ dst, float* __restrict__ deg, int E) {
    int e = blockIdx.x * blockDim.x + threadIdx.x;
    if (e < E) atomicAdd(&deg[dst[e]], 1.0f);
}

__global__ void k_rsqrt(float* p, int n) {
    int i = blockIdx.x * blockDim.x + threadIdx.x;
    if (i < n) p[i] = rsqrtf(p[i]);    // deg >= 1 always (self-loops)
}

// ---------------- f32 WMMA GEMM: C[M,N] = A[M,K] @ B[K,N] (+bias) ----------------
// 2x2 register blocking: one wave computes a 32x32 output tile (4 WMMAs / K-step,
// 2 A-frags + 2 B-frags per step instead of 1A+1B per single WMMA).
// block (32,4): 4 waves cover 128 columns of N; grid = (ceil(M/32), N/128)
__global__ __launch_bounds__(128)
void k_gemm_wmma(const float* __restrict__ A, const float* __restrict__ B,
                 const float* __restrict__ bias, float* __restrict__ C,
                 int M, int K, int N) {
    const int lane = threadIdx.x;                    // 0..31 (wave32)
    const int half = lane >> 4;                      // 0: K pair {0,1}, 1: K pair {2,3}
    const int l15  = lane & 15;
    const int m0   = blockIdx.x * 32;
    const int n0   = (blockIdx.y * blockDim.y + threadIdx.y) * 32;
    if (m0 >= M || n0 >= N) return;                  // wave-uniform

    // A fragment rows (clamped for the ragged last M-tile; stores are guarded)
    const int mr0 = m0 + l15;
    const int mr1 = m0 + 16 + l15;
    const int mc0 = mr0 < M ? mr0 : M - 1;
    const int mc1 = mr1 < M ? mr1 : M - 1;
    const int nc0 = n0 + l15;
    const int nc1 = n0 + 16 + l15;

    const float* Ap0 = A + (size_t)mc0 * K + 2 * half;   // A[m][k + 2*half + {0,1}]
    const float* Ap1 = A + (size_t)mc1 * K + 2 * half;

    v8f acc00 = {}, acc01 = {}, acc10 = {}, acc11 = {};

    for (int k0 = 0; k0 < K; k0 += 4) {
        v2f a0, a1, b0, b1;
        a0.x = Ap0[k0];
        a0.y = Ap0[k0 + 1];
        a1.x = Ap1[k0];
        a1.y = Ap1[k0 + 1];
        const float* Bp = B + (size_t)(k0 + 2 * half) * N;
        b0.x = Bp[nc0];
        b0.y = Bp[N + nc0];
        b1.x = Bp[nc1];
        b1.y = Bp[N + nc1];
        acc00 = __builtin_amdgcn_wmma_f32_16x16x4_f32(false, a0, false, b0,
                                                      (short)0, acc00, false, false);
        acc01 = __builtin_amdgcn_wmma_f32_16x16x4_f32(false, a0, false, b1,
                                                      (short)0, acc01, false, false);
        acc10 = __builtin_amdgcn_wmma_f32_16x16x4_f32(false, a1, false, b0,
                                                      (short)0, acc10, false, false);
        acc11 = __builtin_amdgcn_wmma_f32_16x16x4_f32(false, a1, false, b1,
                                                      (short)0, acc11, false, false);
    }

    // C layout per tile: lane group half selects M offset 0/8, N = n-tile base + l15
    const float bc0 = bias ? bias[nc0] : 0.0f;
    const float bc1 = bias ? bias[nc1] : 0.0f;
    #pragma unroll
    for (int r = 0; r < 8; ++r) {
        int row0 = m0 + r + 8 * half;
        int row1 = row0 + 16;
        if (row0 < M) {
            C[(size_t)row0 * N + nc0] = acc00[r] + bc0;
            C[(size_t)row0 * N + nc1] = acc01[r] + bc1;
        }
        if (row1 < M) {
            C[(size_t)row1 * N + nc0] = acc10[r] + bc0;
            C[(size_t)row1 * N + nc1] = acc11[r] + bc1;
        }
    }
}

// ---------------- propagate: init with self-loop term + bias ----------------
__global__ void k_self_init(const float* __restrict__ h, const float* __restrict__ dis,
                            const float* __restrict__ bias, float* __restrict__ out,
                            long long total) {
    long long idx = (long long)blockIdx.x * blockDim.x + threadIdx.x;
    if (idx >= total) return;
    int i = (int)(idx >> 8);                         // HID = 256
    int f = (int)(idx & (HID - 1));
    float d = dis[i];
    out[idx] = h[idx] * d * d + bias[f];
}

// ---------------- propagate: edge scatter-add (64 threads/edge, float4 each) ----------------
__global__ void k_edge_agg(const long long* __restrict__ src, const long long* __restrict__ dst,
                           const float* __restrict__ dis, const float* __restrict__ h,
                           float* __restrict__ out, int E) {
    long long gid = (long long)blockIdx.x * blockDim.x + threadIdx.x;
    long long total = (long long)E * (HID / 4);
    if (gid >= total) return;
    int e = (int)(gid >> 6);                         // HID/4 = 64 groups per edge
    int g = (int)(gid & 63);
    long long s = src[e];
    long long d = dst[e];
    float nrm = dis[s] * dis[d];
    const float4 v = ((const float4*)(h + (size_t)s * HID))[g];
    float* op = out + (size_t)d * HID + g * 4;
    atomicAdd(op + 0, v.x * nrm);
    atomicAdd(op + 1, v.y * nrm);
    atomicAdd(op + 2, v.z * nrm);
    atomicAdd(op + 3, v.w * nrm);
}

// ---------------- batchnorm ----------------
__global__ void k_zero(float* p, int n) {
    int i = blockIdx.x * blockDim.x + threadIdx.x;
    if (i < n) p[i] = 0.0f;
}

// blockDim = 256 (= HID): thread t owns feature column t over a row-chunk
__global__ void k_bn_accum(const float* __restrict__ X, float* __restrict__ sums, int Nn) {
    int f = threadIdx.x;
    float s = 0.0f, q = 0.0f;
    for (int r = blockIdx.x; r < Nn; r += gridDim.x) {
        float v = X[(size_t)r * HID + f];
        s += v;
        q += v * v;
    }
    atomicAdd(&sums[f], s);
    atomicAdd(&sums[HID + f], q);
}

__global__ void k_bn_fin(const float* __restrict__ sums, const float* __restrict__ gamma,
                         const float* __restrict__ beta, float* __restrict__ sc, int Nn) {
    int f = threadIdx.x;
    float inv_n = 1.0f / (float)Nn;
    float mean = sums[f] * inv_n;
    float var  = sums[HID + f] * inv_n - mean * mean;   // biased variance
    float s    = gamma[f] * rsqrtf(var + BN_EPS);
    sc[f]       = s;                                    // scale
    sc[HID + f] = beta[f] - mean * s;                   // shift
}

__global__ void k_bn_apply(float* __restrict__ X, const float* __restrict__ sc, long long total) {
    long long idx = (long long)blockIdx.x * blockDim.x + threadIdx.x;
    if (idx >= total) return;
    int f = (int)(idx & (HID - 1));
    X[idx] = X[idx] * sc[f] + sc[HID + f];
}

extern "C" void kernel_launch(void* const* d_in, const int* in_sizes, int n_in,
                              void* d_out, int out_size, void* d_ws, size_t ws_size,
                              hipStream_t stream) {
    const float*     x   = (const float*)d_in[0];
    const long long* ei  = (const long long*)d_in[1];   // int64 [2, E]
    const float*     W1  = (const float*)d_in[2];
    const float*     b1  = (const float*)d_in[3];
    const float*     g1  = (const float*)d_in[4];
    const float*     be1 = (const float*)d_in[5];
    const float*     W2  = (const float*)d_in[6];
    const float*     b2  = (const float*)d_in[7];
    const float*     g2  = (const float*)d_in[8];
    const float*     be2 = (const float*)d_in[9];
    const float*     Wfc = (const float*)d_in[10];
    const float*     bfc = (const float*)d_in[11];
    float*           out = (float*)d_out;

    const int F  = 128;
    const int H  = HID;
    const int Nn = in_sizes[0] / F;                  // 50000 nodes
    const int E  = in_sizes[1] / 2;                  // 800000 edges
    const long long* src = ei;
    const long long* dst = ei + E;

    // workspace layout
    float* dis   = (float*)d_ws;                                 // Nn floats
    float* bufH  = dis + (((size_t)Nn + 255) & ~(size_t)255);    // Nn*H (GEMM out)
    float* bufG  = bufH + (size_t)Nn * H;                        // Nn*H (aggregated)
    float* stats = bufG + (size_t)Nn * H;                        // 4*H: sum,sumsq,scale,shift
    float* sums  = stats;
    float* sc    = stats + 2 * H;

    const long long tot   = (long long)Nn * H;                   // 12.8M
    const int  blkEl      = (int)((tot + 255) / 256);
    const int  blkEdge    = (E + 255) / 256;
    const long long eg    = (long long)E * (H / 4);
    const int  blkEg      = (int)((eg + 255) / 256);
    dim3 gGemm((Nn + 31) / 32, H / 128);             // 32-row x 128-col blocks
    dim3 bGemm(32, 4);

    // degree -> deg^{-1/2}
    k_fill1<<<(Nn + 255) / 256, 256, 0, stream>>>(dis, Nn);
    k_deg_edges<<<blkEdge, 256, 0, stream>>>(dst, dis, E);
    k_rsqrt<<<(Nn + 255) / 256, 256, 0, stream>>>(dis, Nn);

    // ---- layer 1: GCNConv(x, W1, b1) + BN(g1, be1) ----
    k_gemm_wmma<<<gGemm, bGemm, 0, stream>>>(x, W1, nullptr, bufH, Nn, F, H);
    k_self_init<<<blkEl, 256, 0, stream>>>(bufH, dis, b1, bufG, tot);
    k_edge_agg<<<blkEg, 256, 0, stream>>>(src, dst, dis, bufH, bufG, E);
    k_zero<<<2, 256, 0, stream>>>(sums, 2 * H);
    k_bn_accum<<<512, 256, 0, stream>>>(bufG, sums, Nn);
    k_bn_fin<<<1, 256, 0, stream>>>(sums, g1, be1, sc, Nn);
    k_bn_apply<<<blkEl, 256, 0, stream>>>(bufG, sc, tot);

    // ---- layer 2: GCNConv(h, W2, b2) + BN(g2, be2) ----
    k_gemm_wmma<<<gGemm, bGemm, 0, stream>>>(bufG, W2, nullptr, bufH, Nn, H, H);
    k_self_init<<<blkEl, 256, 0, stream>>>(bufH, dis, b2, bufG, tot);
    k_edge_agg<<<blkEg, 256, 0, stream>>>(src, dst, dis, bufH, bufG, E);
    k_zero<<<2, 256, 0, stream>>>(sums, 2 * H);
    k_bn_accum<<<512, 256, 0, stream>>>(bufG, sums, Nn);
    k_bn_fin<<<1, 256, 0, stream>>>(sums, g2, be2, sc, Nn);
    k_bn_apply<<<blkEl, 256, 0, stream>>>(bufG, sc, tot);

    // ---- final fc: out = h @ Wfc + bfc ----
    k_gemm_wmma<<<gGemm, bGemm, 0, stream>>>(bufG, Wfc, bfc, out, Nn, H, H);
}